// PatchLayer_21500606284055
// MI455X (gfx1250) — compile-verified
//
#include <hip/hip_runtime.h>
#include <math.h>

// Problem geometry (fixed by the reference)
#define IH    224
#define IW    224
#define IC    3
#define PADP  10
#define PH    244          // padded height/width = 224 + 2*10
#define OH    213          // output positions per axis = 224+20-32+1
#define OW    213
#define LPOS  45369        // OH*OW
#define KH    32
#define KW    32
#define DDIM  3072         // IC*KH*KW, ordered (c, ky, kx) like torch Unfold
#define NPAT  256
#define EPSV  1e-8f

#define IMG_LDS 256        // padded image row buffer (needs >= 247)
#define W_LDS   224        // padded weight row buffer (needs >= 216)

typedef float v2f __attribute__((ext_vector_type(2)));
typedef float v8f __attribute__((ext_vector_type(8)));

// Zero-padded image access: imgP[y][x][c], y,x in padded coords
__device__ __forceinline__ float imgP(const float* __restrict__ img,
                                      int y, int x, int c) {
  if (y < PADP || y >= PADP + IH || x < PADP || x >= PADP + IW) return 0.f;
  return img[((y - PADP) * IW + (x - PADP)) * IC + c];
}

// ---------------------------------------------------------------------------
// Kernel A: Q[y][x] = sum_c imgP(y,x,c)^2 over the padded grid.
// Also zeroes the S accumulator (kernel C accumulates with atomics, and the
// harness replays the launch, so S must be re-zeroed every call).
// ---------------------------------------------------------------------------
__global__ void k_sq(const float* __restrict__ img,
                     float* __restrict__ Q,
                     float* __restrict__ S) {
  int t = blockIdx.x * blockDim.x + threadIdx.x;
  if (t < DDIM) S[t] = 0.f;
  if (t >= PH * PH) return;
  int y = t / PH, x = t % PH;
  float s = 0.f;
  if (y >= PADP && y < PADP + IH && x >= PADP && x < PADP + IW) {
    const float* p = &img[((y - PADP) * IW + (x - PADP)) * IC];
    float a = p[0], b = p[1], c = p[2];
    s = a * a + b * b + c * c;
  }
  Q[t] = s;
}

// ---------------------------------------------------------------------------
// Kernel B1: horizontal 32-tap box sum.  rowsum[y][ox] = sum_{kx<32} Q[y][ox+kx]
// ---------------------------------------------------------------------------
__global__ void k_rowsum(const float* __restrict__ Q,
                         float* __restrict__ rowsum) {
  int t = blockIdx.x * blockDim.x + threadIdx.x;
  if (t >= PH * OW) return;
  int y = t / OW, ox = t % OW;
  const float* q = &Q[y * PH + ox];
  float s = 0.f;
#pragma unroll
  for (int k = 0; k < KW; ++k) s += q[k];
  rowsum[y * OW + ox] = s;
}

// ---------------------------------------------------------------------------
// Kernel B2: vertical 32-tap box sum -> ||un||^2, then W = 1/max(||un||, eps)
// ---------------------------------------------------------------------------
__global__ void k_wmap(const float* __restrict__ rowsum,
                       float* __restrict__ Wv) {
  int t = blockIdx.x * blockDim.x + threadIdx.x;
  if (t >= LPOS) return;
  int oy = t / OW, ox = t % OW;
  float s = 0.f;
#pragma unroll
  for (int k = 0; k < KH; ++k) s += rowsum[(oy + k) * OW + ox];
  Wv[t] = 1.f / fmaxf(sqrtf(s), EPSV);
}

// ---------------------------------------------------------------------------
// Kernel C: S[c*1024 + ky*32 + kx] = sum_{oy,ox} W[oy][ox]*imgP[oy+ky][ox+kx][c]
// via V_WMMA_F32_16X16X4_F32 (fp32 matrix pipe, no precision loss).
//
// Each of the 96 blocks owns one (c, ky); its 8 waves stride the 213 oy rows.
// Per oy, a wave stages the zero-padded image row and W row into its private
// LDS buffers (same-wave DS ordering, no barriers), then runs an unconditional
// 54-iteration Hankel x broadcast-W WMMA loop.
//
// Per-lane VGPR placement (ISA 7.12.2, 32-bit A 16x4): lane m = lane&15,
// element j holds K = j + 2*(lane>=16).  B (4x16) mirrors the C/D half-wave
// split: element j holds row k = j + 2*(lane>=16), column n = lane&15.
// A rows are broadcast (all m identical), so D rows are identical; row 0
// (acc[0], lanes 0..15) carries the 16 partial sums.
// ---------------------------------------------------------------------------
__global__ void k_S_wmma(const float* __restrict__ img,
                         const float* __restrict__ Wv,
                         float* __restrict__ S) {
  __shared__ float ldsImg[8][IMG_LDS];
  __shared__ float ldsW[8][W_LDS];

  const int cky  = blockIdx.x;          // 0..95
  const int c    = cky >> 5;            // channel 0..2
  const int ky   = cky & 31;            // kernel row 0..31
  const int lane = threadIdx.x & 31;
  const int wave = threadIdx.x >> 5;    // 8 waves per block
  const int half = lane >> 4;           // 0: lanes 0-15, 1: lanes 16-31
  const int n    = lane & 15;           // tile column
  const int kb   = half * 2;            // K base for this half-wave

  float* im = ldsImg[wave];
  float* wr = ldsW[wave];

  v8f acc0 = {0.f, 0.f, 0.f, 0.f, 0.f, 0.f, 0.f, 0.f};  // kx 0..15
  v8f acc1 = {0.f, 0.f, 0.f, 0.f, 0.f, 0.f, 0.f, 0.f};  // kx 16..31

  for (int oy = wave; oy < OH; oy += 8) {
    const int r = oy + ky;                       // padded image row
    const float* wrow = &Wv[oy * OW];
    // Stage zero-padded rows (uniform trip counts; all guards live here)
#pragma unroll
    for (int i = lane; i < IMG_LDS; i += 32) im[i] = imgP(img, r, i, c);
#pragma unroll
    for (int i = lane; i < W_LDS; i += 32) wr[i] = (i < OW) ? wrow[i] : 0.f;

    // Unconditional WMMA sweep over ox chunks of K=4 (216 covers 213 + pad)
    for (int ox0 = 0; ox0 < 216; ox0 += 4) {
      const int k0 = ox0 + kb;
      v2f a, b0, b1;
      a.x  = wr[k0];
      a.y  = wr[k0 + 1];
      b0.x = im[k0 + n];
      b0.y = im[k0 + 1 + n];
      b1.x = im[k0 + 16 + n];
      b1.y = im[k0 + 17 + n];
      acc0 = __builtin_amdgcn_wmma_f32_16x16x4_f32(false, a, false, b0,
                                                   (short)0, acc0, false, false);
      acc1 = __builtin_amdgcn_wmma_f32_16x16x4_f32(false, a, false, b1,
                                                   (short)0, acc1, false, false);
    }
  }

  if (half == 0) {  // row M=0 lives in element 0 of lanes 0..15
    const int base = (c * KH + ky) * KW;
    atomicAdd(&S[base + n],      acc0[0]);
    atomicAdd(&S[base + 16 + n], acc1[0]);
  }
}

// ---------------------------------------------------------------------------
// Kernel D: one wave per patch.
// out[p] = 1 - (S . patch_p) / (L * max(||patch_p||, eps))
// ---------------------------------------------------------------------------
__global__ void k_out(const float* __restrict__ S,
                      const float* __restrict__ patches,
                      float* __restrict__ out) {
  const int wave = threadIdx.x >> 5;
  const int lane = threadIdx.x & 31;
  const int p = blockIdx.x * (blockDim.x >> 5) + wave;
  if (p >= NPAT) return;

  const float4* pr = (const float4*)&patches[(size_t)p * DDIM];
  const float4* s4 = (const float4*)S;
  float dot = 0.f, nsq = 0.f;
  for (int j = lane; j < DDIM / 4; j += 32) {
    float4 v = pr[j], s = s4[j];
    dot += v.x * s.x + v.y * s.y + v.z * s.z + v.w * s.w;
    nsq += v.x * v.x + v.y * v.y + v.z * v.z + v.w * v.w;
  }
#pragma unroll
  for (int off = 16; off > 0; off >>= 1) {
    dot += __shfl_xor(dot, off, 32);
    nsq += __shfl_xor(nsq, off, 32);
  }
  if (lane == 0)
    out[p] = 1.f - dot / ((float)LPOS * fmaxf(sqrtf(nsq), EPSV));
}

// ---------------------------------------------------------------------------
extern "C" void kernel_launch(void* const* d_in, const int* in_sizes, int n_in,
                              void* d_out, int out_size, void* d_ws, size_t ws_size,
                              hipStream_t stream) {
  const float* img     = (const float*)d_in[0];   // [224,224,3] fp32
  const float* patches = (const float*)d_in[1];   // [256,3072]  fp32
  float* out = (float*)d_out;                     // [256] fp32

  float* ws     = (float*)d_ws;
  float* Q      = ws;                    // PH*PH   = 59536
  float* rowsum = Q + PH * PH;           // PH*OW   = 51972
  float* Wv     = rowsum + PH * OW;      // LPOS    = 45369
  float* S      = Wv + LPOS;             // DDIM    = 3072   (~640 KB total)

  k_sq    <<<(PH * PH + 255) / 256, 256, 0, stream>>>(img, Q, S);
  k_rowsum<<<(PH * OW + 255) / 256, 256, 0, stream>>>(Q, rowsum);
  k_wmap  <<<(LPOS  + 255) / 256, 256, 0, stream>>>(rowsum, Wv);
  k_S_wmma<<<IC * KH, 256, 0, stream>>>(img, Wv, S);   // 96 blocks x 8 waves
  k_out   <<<NPAT / 8, 256, 0, stream>>>(S, patches, out);
  (void)in_sizes; (void)n_in; (void)out_size; (void)ws_size;
}